// GraphConv_46093589020994
// MI455X (gfx1250) — compile-verified
//
#include <hip/hip_runtime.h>
#include <math.h>

#define NE    8192
#define NU    8192
#define DD    64
#define NR    24
#define NEDGE 524288
#define NNZS  262144
#define TK    10
#define SIMW  4          // waves per simtopk workgroup

typedef __attribute__((ext_vector_type(2))) float v2f;
typedef __attribute__((ext_vector_type(8))) float v8f;

__device__ __forceinline__ bool better(float v, int i, float v2, int i2) {
    return (v > v2) || (v == v2 && i < i2);
}

// ---------------------------------------------------------------- init / zero
__global__ void k_init(const float* __restrict__ ue, const float* __restrict__ ee,
                       float* ucur, float* usum, float* ecur, float* esum, int n) {
    int i = blockIdx.x * blockDim.x + threadIdx.x;
    if (i < n) {
        float u = ue[i], e = ee[i];
        ucur[i] = u; usum[i] = u;      // res lists start with the RAW embeddings
        ecur[i] = e; esum[i] = e;
    }
}

__global__ void k_zero_hop(float* msum, float* uagg, unsigned* m, float* denom,
                           float* cnt, int nbig, int nsmall) {
    int i = blockIdx.x * blockDim.x + threadIdx.x;
    if (i < nbig)   { msum[i] = 0.f; uagg[i] = 0.f; }
    if (i < nsmall) { m[i] = 0u; denom[i] = 0.f; cnt[i] = 0.f; }
}

// ---------------------------------------------------------------- edge pass 1
// att[e] = ||h*er||^2 * ||t*er||^2 ; segment-max over head (uint atomicMax, att>=0)
__global__ void k_edge_att(const float* __restrict__ ecur, const float* __restrict__ rel,
                           const int* __restrict__ head, const int* __restrict__ tail,
                           const int* __restrict__ etype, float* att, unsigned* m) {
    int gt = blockIdx.x * 256 + threadIdx.x;
    int e = gt >> 4, l = gt & 15;
    if (e >= NEDGE) return;
    int h = head[e], t = tail[e], r = etype[e];
    int d0 = l * 4;
    float4 h4 = *(const float4*)(ecur + (size_t)h * DD + d0);
    float4 t4 = *(const float4*)(ecur + (size_t)t * DD + d0);
    float4 r4 = *(const float4*)(rel + (size_t)r * DD + d0);
    float hx = h4.x * r4.x, hy = h4.y * r4.y, hz = h4.z * r4.z, hw = h4.w * r4.w;
    float tx = t4.x * r4.x, ty = t4.y * r4.y, tz = t4.z * r4.z, tw = t4.w * r4.w;
    float sh = hx * hx + hy * hy + hz * hz + hw * hw;
    float st = tx * tx + ty * ty + tz * tz + tw * tw;
    #pragma unroll
    for (int mk = 8; mk >= 1; mk >>= 1) {      // masks <16 stay within the 16-lane group
        sh += __shfl_xor(sh, mk, 32);
        st += __shfl_xor(st, mk, 32);
    }
    if (l == 0) {
        float a = sh * st;                      // (sqrt(sh)*sqrt(st))^2 == sh*st
        att[e] = a;
        atomicMax(m + h, __float_as_uint(a));   // valid: a >= 0
    }
}

// ---------------------------------------------------------------- edge pass 2
__global__ void k_edge_exp(const int* __restrict__ head, const float* __restrict__ att,
                           const unsigned* __restrict__ m, float* eexp,
                           float* denom, float* cnt) {
    int e = blockIdx.x * blockDim.x + threadIdx.x;
    if (e >= NEDGE) return;
    int h = head[e];
    float ex = expf(att[e] - __uint_as_float(m[h]));
    eexp[e] = ex;
    atomicAdd(denom + h, ex);
    atomicAdd(cnt + h, 1.0f);
}

// ---------------------------------------------------------------- edge pass 3
__global__ void k_edge_msg(const float* __restrict__ ecur, const float* __restrict__ rel,
                           const int* __restrict__ head, const int* __restrict__ tail,
                           const int* __restrict__ etype, const float* __restrict__ eexp,
                           const float* __restrict__ denom, float* msum) {
    int gt = blockIdx.x * 256 + threadIdx.x;
    int e = gt >> 4, l = gt & 15;
    if (e >= NEDGE) return;
    int h = head[e], t = tail[e], r = etype[e];
    float w = eexp[e] / denom[h];
    int d0 = l * 4;
    float4 t4 = *(const float4*)(ecur + (size_t)t * DD + d0);
    float4 r4 = *(const float4*)(rel + (size_t)r * DD + d0);
    float* dst = msum + (size_t)h * DD + d0;
    atomicAdd(dst + 0, w * t4.x * r4.x);
    atomicAdd(dst + 1, w * t4.y * r4.y);
    atomicAdd(dst + 2, w * t4.z * r4.z);
    atomicAdd(dst + 3, w * t4.w * r4.w);
}

// ---------------------------------------------------------------- user sparse agg
__global__ void k_user_sparse(const float* __restrict__ ecur, const int* __restrict__ irow,
                              const int* __restrict__ icol, const float* __restrict__ ival,
                              float* uagg) {
    int gt = blockIdx.x * 256 + threadIdx.x;
    int n = gt >> 4, l = gt & 15;
    if (n >= NNZS) return;
    int u = irow[n], c = icol[n];
    float v = ival[n];
    int d0 = l * 4;
    float4 e4 = *(const float4*)(ecur + (size_t)c * DD + d0);
    float* dst = uagg + (size_t)u * DD + d0;
    atomicAdd(dst + 0, v * e4.x);
    atomicAdd(dst + 1, v * e4.y);
    atomicAdd(dst + 2, v * e4.z);
    atomicAdd(dst + 3, v * e4.w);
}

// ---------------------------------------------------------------- user relation mix + l2norm
__global__ __launch_bounds__(64)
void k_user_mix(float* ucur, const float* __restrict__ rel, const float* __restrict__ uagg,
                float* usum) {
    __shared__ float s[NR];
    __shared__ float red[DD];
    int u = blockIdx.x, d = threadIdx.x;
    if (d < NR) {
        float acc = 0.f;
        for (int k = 0; k < DD; ++k) acc += ucur[(size_t)u * DD + k] * rel[d * DD + k];
        s[d] = acc;
    }
    __syncthreads();
    if (d == 0) {
        float mx = s[0];
        for (int r = 1; r < NR; ++r) mx = fmaxf(mx, s[r]);
        float sm = 0.f;
        for (int r = 0; r < NR; ++r) { float e = expf(s[r] - mx); s[r] = e; sm += e; }
        float inv = 1.0f / sm;
        for (int r = 0; r < NR; ++r) s[r] *= inv;
    }
    __syncthreads();
    float mix = 0.f;
    for (int r = 0; r < NR; ++r) mix += s[r] * rel[r * DD + d];
    float ua = uagg[(size_t)u * DD + d];
    ua = ua + mix * ua;
    red[d] = ua * ua;
    __syncthreads();
    for (int off = 32; off >= 1; off >>= 1) { if (d < off) red[d] += red[d + off]; __syncthreads(); }
    float nv = ua / fmaxf(sqrtf(red[0]), 1e-12f);
    ucur[(size_t)u * DD + d] = nv;
    usum[(size_t)u * DD + d] += nv;
}

// ---------------------------------------------------------------- entity finalize (mean + l2norm)
__global__ __launch_bounds__(64)
void k_entity_fin(const float* __restrict__ msum, const float* __restrict__ cnt,
                  float* ecur, float* esum) {
    __shared__ float red[DD];
    int i = blockIdx.x, d = threadIdx.x;
    float v = msum[(size_t)i * DD + d] / fmaxf(cnt[i], 1.0f);
    red[d] = v * v;
    __syncthreads();
    for (int off = 32; off >= 1; off >>= 1) { if (d < off) red[d] += red[d + off]; __syncthreads(); }
    float nv = v / fmaxf(sqrtf(red[0]), 1e-12f);
    ecur[(size_t)i * DD + d] = nv;
    esum[(size_t)i * DD + d] += nv;
}

// ---------------------------------------------------------------- row normalize (build_adj style)
__global__ __launch_bounds__(64)
void k_rownorm(const float* __restrict__ in, float scale, float* mean_out, float* cn) {
    __shared__ float red[DD];
    int i = blockIdx.x, d = threadIdx.x;
    float v = in[(size_t)i * DD + d] * scale;
    if (mean_out) mean_out[(size_t)i * DD + d] = v;
    red[d] = v * v;
    __syncthreads();
    for (int off = 32; off >= 1; off >>= 1) { if (d < off) red[d] += red[d + off]; __syncthreads(); }
    cn[(size_t)i * DD + d] = v / sqrtf(red[0]);   // no eps, matches reference
}

__global__ void k_user_res(const float* __restrict__ usum, float* out, int n) {
    int i = blockIdx.x * blockDim.x + threadIdx.x;
    if (i < n) out[i] = usum[i] * (1.0f / 3.0f);
}

// ---------------------------------------------------------------- fused sim GEMM (WMMA f32) + top-k
// 4 waves per 16-row strip; each wave strides over column tiles with a private LDS
// tile region (no in-loop barrier: same-wave DS ops are in-order, compiler inserts
// s_wait_dscnt for the store->load dependence). 2048 waves total hide the dependent
// v_wmma_f32_16x16x4_f32 accumulation-chain latency via SIMD multithreading.
__global__ __launch_bounds__(32 * SIMW)
void k_simtopk(const float* __restrict__ cn, float* __restrict__ tkv, int* __restrict__ tki) {
    __shared__ float tile[SIMW][16 * 16];
    __shared__ float mv[16][SIMW * 20];
    __shared__ int   mi[16][SIMW * 20];
    const int tid  = threadIdx.x;
    const int wave = tid >> 5;
    const int lane = tid & 31;
    const int row0 = blockIdx.x * 16;
    const int rsel = lane & 15;          // M (for A) / N (for B) index held by this lane
    const int koff = (lane >> 4) * 2;    // k sub-offset per WMMA layout
    const int half = lane >> 4;

    // Preload A fragments (invariant over column tiles): 16 k-steps of 16x4 fp32
    v2f a[16];
    #pragma unroll
    for (int j = 0; j < 16; ++j) {
        const float* p = cn + (size_t)(row0 + rsel) * DD + j * 4 + koff;
        a[j] = v2f{p[0], p[1]};
    }

    // Per-lane sorted top-10 (lane handles row lane&15, column half lane>>4)
    float tv[TK]; int ti[TK];
    #pragma unroll
    for (int k = 0; k < TK; ++k) { tv[k] = -INFINITY; ti[k] = 0x7fffffff; }

    for (int ct = wave; ct < NE / 16; ct += SIMW) {
        const int col0 = ct * 16;
        v8f acc = {0.f, 0.f, 0.f, 0.f, 0.f, 0.f, 0.f, 0.f};
        #pragma unroll
        for (int j = 0; j < 16; ++j) {
            const float* p = cn + (size_t)(col0 + rsel) * DD + j * 4 + koff;
            v2f b = v2f{p[0], p[1]};
            acc = __builtin_amdgcn_wmma_f32_16x16x4_f32(false, a[j], false, b,
                                                        (short)0, acc, false, false);
        }
        // C layout: VGPR v, lane l -> row = v + 8*(l>>4), col = l&15
        #pragma unroll
        for (int v = 0; v < 8; ++v)
            tile[wave][(v + (half << 3)) * 16 + rsel] = acc[v];

        const int cbase = half * 8;
        #pragma unroll
        for (int cc = 0; cc < 8; ++cc) {
            float val = tile[wave][rsel * 16 + cbase + cc];
            int gidx = col0 + cbase + cc;
            if (better(val, gidx, tv[TK - 1], ti[TK - 1])) {
                float v0 = val; int i0 = gidx;
                #pragma unroll
                for (int j = 0; j < TK; ++j) {
                    bool b = better(v0, i0, tv[j], ti[j]);
                    float ov = tv[j]; int oi = ti[j];
                    if (b) { tv[j] = v0; ti[j] = i0; v0 = ov; i0 = oi; }
                }
            }
        }
    }

    // Merge SIMW*2 half-column lists per row, emit sorted top-10
    for (int k = 0; k < TK; ++k) {
        mv[rsel][wave * 20 + half * TK + k] = tv[k];
        mi[rsel][wave * 20 + half * TK + k] = ti[k];
    }
    __syncthreads();
    if (tid < 16) {
        for (int k = 0; k < TK; ++k) {
            float best = -INFINITY; int bi = 0x7fffffff; int bs = -1;
            for (int s = 0; s < SIMW * 20; ++s) {
                float v = mv[tid][s]; int id = mi[tid][s];
                if (better(v, id, best, bi)) { best = v; bi = id; bs = s; }
            }
            mv[tid][bs] = -INFINITY;
            mi[tid][bs] = 0x7fffffff;
            tkv[(size_t)(row0 + tid) * TK + k] = best;
            tki[(size_t)(row0 + tid) * TK + k] = bi;
        }
    }
}

__global__ void k_rowsum(const float* __restrict__ tkv, float* dinv) {
    int i = blockIdx.x * blockDim.x + threadIdx.x;
    if (i >= NE) return;
    float s = 0.f;
    for (int k = 0; k < TK; ++k) s += tkv[(size_t)i * TK + k];
    dinv[i] = 1.0f / sqrtf(s);
}

__global__ void k_adj_zero(float4* adj, size_t n4) {
    size_t i = (size_t)blockIdx.x * blockDim.x + threadIdx.x;
    if (i < n4) adj[i] = make_float4(0.f, 0.f, 0.f, 0.f);
}

__global__ void k_adj_scatter(const float* __restrict__ tkvO, const int* __restrict__ tkiO,
                              const float* __restrict__ dO,
                              const float* __restrict__ tkvR, const int* __restrict__ tkiR,
                              const float* __restrict__ dR, float* adj) {
    int i = blockIdx.x * blockDim.x + threadIdx.x;
    if (i >= NE) return;
    float diO = dO[i], diR = dR[i];
    float* row = adj + (size_t)i * NE;
    for (int k = 0; k < TK; ++k) {         // (1-LAMBDA) = 0.5 part (res adj)
        int j = tkiR[(size_t)i * TK + k];
        row[j] += 0.5f * tkvR[(size_t)i * TK + k] * diR * dR[j];
    }
    for (int k = 0; k < TK; ++k) {         // LAMBDA = 0.5 part (origin adj); same-thread RMW
        int j = tkiO[(size_t)i * TK + k];
        row[j] += 0.5f * tkvO[(size_t)i * TK + k] * diO * dO[j];
    }
}

// ---------------------------------------------------------------- launcher
extern "C" void kernel_launch(void* const* d_in, const int* in_sizes, int n_in,
                              void* d_out, int out_size, void* d_ws, size_t ws_size,
                              hipStream_t stream) {
    const float* ue   = (const float*)d_in[0];
    const float* ee   = (const float*)d_in[1];
    const float* rel  = (const float*)d_in[2];
    const int* eidx   = (const int*)d_in[3];
    const int* etype  = (const int*)d_in[4];
    const int* iidx   = (const int*)d_in[5];
    const float* ival = (const float*)d_in[6];
    const int* head = eidx;
    const int* tail = eidx + NEDGE;
    const int* irow = iidx;
    const int* icol = iidx + NNZS;

    float* w = (float*)d_ws;
    size_t o = 0;
    float* ecur = w + o; o += (size_t)NE * DD;
    float* esum = w + o; o += (size_t)NE * DD;
    float* ucur = w + o; o += (size_t)NU * DD;
    float* usum = w + o; o += (size_t)NU * DD;
    float* msum = w + o; o += (size_t)NE * DD;
    float* uagg = w + o; o += (size_t)NU * DD;
    float* cnO  = w + o; o += (size_t)NE * DD;
    float* cnR  = w + o; o += (size_t)NE * DD;
    float* att  = w + o; o += NEDGE;
    float* eexp = w + o; o += NEDGE;
    unsigned* mmax = (unsigned*)(w + o); o += NE;
    float* denom = w + o; o += NE;
    float* cnt   = w + o; o += NE;
    float* tkvO  = w + o; o += NE * TK;
    float* tkvR  = w + o; o += NE * TK;
    int* tkiO = (int*)(w + o); o += NE * TK;
    int* tkiR = (int*)(w + o); o += NE * TK;
    float* dO = w + o; o += NE;
    float* dR = w + o; o += NE;

    float* out_e   = (float*)d_out;
    float* out_u   = out_e + (size_t)NE * DD;
    float* out_adj = out_u + (size_t)NU * DD;

    k_init<<<(NE * DD + 255) / 256, 256, 0, stream>>>(ue, ee, ucur, usum, ecur, esum, NE * DD);

    for (int hop = 0; hop < 2; ++hop) {
        k_zero_hop<<<(NE * DD + 255) / 256, 256, 0, stream>>>(msum, uagg, mmax, denom, cnt,
                                                              NE * DD, NE);
        k_edge_att<<<NEDGE * 16 / 256, 256, 0, stream>>>(ecur, rel, head, tail, etype, att, mmax);
        k_edge_exp<<<NEDGE / 256, 256, 0, stream>>>(head, att, mmax, eexp, denom, cnt);
        k_edge_msg<<<NEDGE * 16 / 256, 256, 0, stream>>>(ecur, rel, head, tail, etype, eexp,
                                                         denom, msum);
        k_user_sparse<<<NNZS * 16 / 256, 256, 0, stream>>>(ecur, irow, icol, ival, uagg);
        k_user_mix<<<NU, 64, 0, stream>>>(ucur, rel, uagg, usum);
        k_entity_fin<<<NE, 64, 0, stream>>>(msum, cnt, ecur, esum);
    }

    k_rownorm<<<NE, 64, 0, stream>>>(esum, 1.0f / 3.0f, out_e, cnR);   // entity_res_emb + cn
    k_rownorm<<<NE, 64, 0, stream>>>(ee, 1.0f, nullptr, cnO);          // origin cn
    k_user_res<<<(NU * DD + 255) / 256, 256, 0, stream>>>(usum, out_u, NU * DD);

    k_simtopk<<<NE / 16, 32 * SIMW, 0, stream>>>(cnO, tkvO, tkiO);
    k_simtopk<<<NE / 16, 32 * SIMW, 0, stream>>>(cnR, tkvR, tkiR);
    k_rowsum<<<(NE + 255) / 256, 256, 0, stream>>>(tkvO, dO);
    k_rowsum<<<(NE + 255) / 256, 256, 0, stream>>>(tkvR, dR);

    size_t n4 = (size_t)NE * NE / 4;
    k_adj_zero<<<(unsigned)((n4 + 255) / 256), 256, 0, stream>>>((float4*)out_adj, n4);
    k_adj_scatter<<<(NE + 255) / 256, 256, 0, stream>>>(tkvO, tkiO, dO, tkvR, tkiR, dR, out_adj);
}